// tel_mpo_38577396253038
// MI455X (gfx1250) — compile-verified
//
#include <hip/hip_runtime.h>

// MPO measurement-string contraction for MI455X (gfx1250, wave32).
//
// Complex 16x8 @ 8x8 chain embedded as real 16x16x16 matmuls via the 2x2
// real-block trick: psi_rr = [R|I] (16x16), W = [[Br,Bi],[-Bi,Br]] (16x16),
// so each site is 4 chained v_wmma_f32_16x16x4_f32. One wave per batch
// element. Deterministic 0.25/site rescale keeps f32 in range and cancels in
// the trace normalization. D->A relayout via per-wave LDS. Measurement bits
// live in a wave-uniform 64-bit SGPR mask (built with __ballot): per-step
// block selection is pure SALU.
//
// Fast path: a prep kernel pre-swizzles all 120 (site,m) blocks into
// B-operand lane layout (f32, sign & scale folded) in d_ws, so the main
// loop fetches B with two global_load_b128 per step instead of
// 8 x (global_load_b64 + cvt_f32_f64 + mul).

typedef __attribute__((ext_vector_type(2))) float v2f;
typedef __attribute__((ext_vector_type(8))) float v8f;

#define NSITES 60            // measured sites (L-2)
#define BATCH 16384
#define WAVES_PER_BLOCK 8    // 256 threads = 8 wave32
#define TABLE_BYTES (NSITES * 2 * 32 * 8 * sizeof(float))   // 122880

// Per-lane B-operand slot mapping (same for every site):
// slot j = 2t+r holds W[K][N], K = 4t + r + 2h, N = lane&15, where
// W = [[Br, Bi], [-Bi, Br]]:
//   part = (K>=8) ^ (N>=8)  (0 -> Re, 1 -> Im), negate iff (K>=8 && N<8).
__device__ __forceinline__ void slot_map(int lane, int* dj, float* sc) {
    const int h = (lane >> 4) & 1;
    const int N = lane & 15;
    #pragma unroll
    for (int j = 0; j < 8; ++j) {
        int t = j >> 1, r = j & 1;
        int K = 4 * t + r + 2 * h;
        int row = K & 7, col = N & 7;
        int part = ((K >= 8) ^ (N >= 8)) ? 1 : 0;
        bool neg = (K >= 8) && (N < 8);
        dj[j] = (row * 8 + col) * 2 + part;
        sc[j] = neg ? -0.25f : 0.25f;
    }
}

// ---- prep: swizzle blocks[1..60][m] into B-operand lane layout ------------
// table[((site-1)*2+m)*256 + lane*8 + j], pair = (site-1)*2+m = 0..119,
// block double base = (pair+2)*128.
__global__ __launch_bounds__(256, 1)
void mpo_prep_kernel(const double* __restrict__ blk, float* __restrict__ table) {
    const int tid  = blockIdx.x * 256 + threadIdx.x;   // 0 .. 3839
    const int pair = tid >> 5;
    const int lane = tid & 31;
    if (pair >= NSITES * 2) return;
    int dj[8]; float sc[8];
    slot_map(lane, dj, sc);
    const double* bb = blk + (long long)(pair + 2) * 128;
    float* tw = table + ((size_t)pair * 32 + lane) * 8;
    #pragma unroll
    for (int j = 0; j < 8; ++j)
        tw[j] = (float)bb[dj[j]] * sc[j];
}

template <bool FAST>
__global__ __launch_bounds__(256, 1)
void mpo_chain_kernel(const double* __restrict__ blk,      // [62][2][8][8] complex128
                      const long long* __restrict__ meas,  // [B][60] int64
                      const float* __restrict__ table,     // pre-swizzled B operands (FAST)
                      double* __restrict__ out)            // [B][4][4] complex128
{
    // Per-wave scratch: 256 f32 for psi relayout, 512 f32 for psiF (Re|Im).
    __shared__ float shPsi[WAVES_PER_BLOCK][256];
    __shared__ float shF[WAVES_PER_BLOCK][512];

    const int lane = threadIdx.x & 31;
    // Force the wave id (hence element id / base addresses) into an SGPR.
    const int wave = __builtin_amdgcn_readfirstlane(threadIdx.x >> 5);
    const int e    = blockIdx.x * WAVES_PER_BLOCK + wave;   // uniform
    const int h    = (lane >> 4) & 1;
    const int n15  = lane & 15;

    int dj[8]; float sc[8];
    slot_map(lane, dj, sc);

    // measurement bits -> wave-uniform 64-bit SGPR mask
    int mlo = (int)meas[(long long)e * NSITES + lane];
    int mhi = (lane + 32 < NSITES) ? (int)meas[(long long)e * NSITES + lane + 32] : 0;
    unsigned long long mask =  (unsigned long long)__ballot(mlo & 1)
                            | ((unsigned long long)__ballot(mhi & 1) << 32);

    // init psi0 directly in A-operand layout:
    // psi0_rr[i][K]: K<8 -> Re psi0[i][K], K>=8 -> Im psi0[i][K-8];
    // psi0[i][j] = blocks[0][i&1][i>>1][j].
    float a[8];
    #pragma unroll
    for (int j = 0; j < 8; ++j) {
        int t = j >> 1, r = j & 1;
        int K = 4 * t + r + 2 * h;
        int i = n15;
        long long cidx = (long long)(i & 1) * 64 + (i >> 1) * 8 + (K & 7);
        a[j] = (float)blk[2 * cidx + (K >> 3)];
    }

    float* myPsi = shPsi[wave];

    // ---- 60-step chain: one 16x16x16 real matmul per site -----------------
    for (int s = 0; s < NSITES; ++s) {
        long long m = (long long)((mask >> s) & 1ull);   // SALU

        float b[8];
        if (FAST) {
            // two b128 loads from the pre-swizzled table
            const float* tb = table + ((size_t)(2 * s) + m) * 256 + lane * 8;
            float4 lo = *(const float4*)tb;
            float4 hi = *(const float4*)(tb + 4);
            b[0] = lo.x; b[1] = lo.y; b[2] = lo.z; b[3] = lo.w;
            b[4] = hi.x; b[5] = hi.y; b[6] = hi.z; b[7] = hi.w;
        } else {
            const double* bb = blk + ((long long)(s + 1) * 256 + m * 128);
            #pragma unroll
            for (int j = 0; j < 8; ++j)
                b[j] = (float)bb[dj[j]] * sc[j];
        }

        v8f acc = {};
        #pragma unroll
        for (int t = 0; t < 4; ++t) {
            v2f At = { a[2 * t], a[2 * t + 1] };
            v2f Bt = { b[2 * t], b[2 * t + 1] };
            acc = __builtin_amdgcn_wmma_f32_16x16x4_f32(
                false, At, false, Bt, (short)0, acc, false, false);
        }

        // D -> A relayout through LDS (same-wave, hardware-ordered).
        #pragma unroll
        for (int v = 0; v < 8; ++v)
            myPsi[(v + 8 * h) * 16 + n15] = acc[v];     // row-major [M][N]
        #pragma unroll
        for (int j = 0; j < 8; ++j) {
            int t = j >> 1, r = j & 1;
            int K = 4 * t + r + 2 * h;
            a[j] = myPsi[n15 * 16 + K];                 // A: (M=n15, K)
        }
    }

    // ---- terminal block: psiF = psi(16x8 cplx) @ last(8x16 cplx) ----------
    // last[r][N] = blocks[61][N>>3][r][N&7]; B_R = [Lr; -Li], B_I = [Li; Lr].
    const double* lb = blk + (long long)61 * 256;
    float bR[8], bI[8];
    #pragma unroll
    for (int j = 0; j < 8; ++j) {
        int t = j >> 1, r = j & 1;
        int K = 4 * t + r + 2 * h;
        int N = n15;
        long long cidx = (long long)(N >> 3) * 64 + (K & 7) * 8 + (N & 7);
        double re = lb[2 * cidx], im = lb[2 * cidx + 1];
        bR[j] = (K >= 8) ? (float)(-im) : (float)re;
        bI[j] = (K >= 8) ? (float)re    : (float)im;
    }
    v8f dR = {}, dI = {};
    #pragma unroll
    for (int t = 0; t < 4; ++t) {
        v2f At  = { a[2 * t],  a[2 * t + 1] };
        v2f BtR = { bR[2 * t], bR[2 * t + 1] };
        v2f BtI = { bI[2 * t], bI[2 * t + 1] };
        dR = __builtin_amdgcn_wmma_f32_16x16x4_f32(
            false, At, false, BtR, (short)0, dR, false, false);
        dI = __builtin_amdgcn_wmma_f32_16x16x4_f32(
            false, At, false, BtI, (short)0, dI, false, false);
    }

    float* shr = shF[wave];
    #pragma unroll
    for (int v = 0; v < 8; ++v) {
        shr[(v + 8 * h) * 16 + n15]       = dR[v];   // Re psiF [16][16]
        shr[256 + (v + 8 * h) * 16 + n15] = dI[v];   // Im psiF
    }

    // ---- rho[p][q] = sum_{a,c} psi2[p,ac] * conj(psi2[q,ac]) --------------
    // psi2[q, a*8+c] = psiF[2a + (q>>1)][(q&1)*8 + c]. Lanes 0..15 own (p,q).
    {
        int p = n15 >> 2, q = n15 & 3;
        float accR = 0.0f, accI = 0.0f;
        #pragma unroll 2
        for (int aa = 0; aa < 8; ++aa) {
            #pragma unroll
            for (int c = 0; c < 8; ++c) {
                int i1 = 2 * aa + (p >> 1), k1 = ((p & 1) << 3) + c;
                int i2 = 2 * aa + (q >> 1), k2 = ((q & 1) << 3) + c;
                float r1 = shr[i1 * 16 + k1], m1 = shr[256 + i1 * 16 + k1];
                float r2 = shr[i2 * 16 + k2], m2 = shr[256 + i2 * 16 + k2];
                accR += r1 * r2 + m1 * m2;     // z1 * conj(z2)
                accI += m1 * r2 - r1 * m2;
            }
        }
        // trace from diagonal lanes (p==q at lanes 0,5,10,15)
        float trR = __shfl(accR, 0) + __shfl(accR, 5) + __shfl(accR, 10) + __shfl(accR, 15);
        float trI = __shfl(accI, 0) + __shfl(accI, 5) + __shfl(accI, 10) + __shfl(accI, 15);

        if (lane < 16) {
            double den = (double)trR * (double)trR + (double)trI * (double)trI;
            double oR  = ((double)accR * trR + (double)accI * trI) / den;
            double oI  = ((double)accI * trR - (double)accR * trI) / den;
            long long o = ((long long)e * 16 + n15) * 2;
            out[o]     = oR;
            out[o + 1] = oI;
        }
    }
}

extern "C" void kernel_launch(void* const* d_in, const int* in_sizes, int n_in,
                              void* d_out, int out_size, void* d_ws, size_t ws_size,
                              hipStream_t stream) {
    (void)in_sizes; (void)n_in; (void)out_size;
    const double*    blocks  = (const double*)d_in[0];     // complex128 [62][2][8][8]
    const long long* measure = (const long long*)d_in[1];  // int64 [16384][60]
    double*          rho     = (double*)d_out;             // complex128 [16384][4][4]

    const int grid = BATCH / WAVES_PER_BLOCK;              // 2048 blocks x 256 thr
    if (d_ws != nullptr && ws_size >= TABLE_BYTES) {
        float* table = (float*)d_ws;
        mpo_prep_kernel<<<(NSITES * 2 * 32 + 255) / 256, 256, 0, stream>>>(blocks, table);
        mpo_chain_kernel<true><<<grid, 256, 0, stream>>>(blocks, measure, table, rho);
    } else {
        mpo_chain_kernel<false><<<grid, 256, 0, stream>>>(blocks, measure, nullptr, rho);
    }
}